// MLGraphConstruction_39015482917560
// MI455X (gfx1250) — compile-verified
//
#include <hip/hip_runtime.h>

typedef __attribute__((ext_vector_type(2))) float v2f;
typedef __attribute__((ext_vector_type(8))) float v8f;

#define NPTS 8192
#define NF   14
#define DEMB 8
#define NH   64
#define KNBR 64
#define NK   (NPTS * KNBR)
#define INF_F __builtin_inff()

// ---------------------------------------------------------------------------
// Kernel A: pack emb (first 8 cols of x) contiguously + squared norms
// ---------------------------------------------------------------------------
__global__ void pack_kernel(const float* __restrict__ x,
                            float* __restrict__ embw,
                            float* __restrict__ sq) {
  int i = blockIdx.x * blockDim.x + threadIdx.x;
  if (i < NPTS) {
    float s = 0.f;
    #pragma unroll
    for (int d = 0; d < DEMB; ++d) {
      float v = x[i * NF + d];
      embw[i * DEMB + d] = v;
      s += v * v;
    }
    sq[i] = s;
  }
}

// ---------------------------------------------------------------------------
// Kernel B: per-row top-64 neighbors.
// One WG (256 thr = 8 waves) handles a 16-row band. Each 4096-col half:
//   WMMA f32 16x16x4 gram tiles -> d2 band in LDS (16x4096 f32 = 256 KB)
//   -> 64-iteration lazy argmin (16 threads per row).
// Two sorted half-lists merged into final sorted 64 neighbors.
// ---------------------------------------------------------------------------
__global__ void topk_kernel(const float* __restrict__ emb,
                            const float* __restrict__ sq,
                            int* __restrict__ nbr) {
  extern __shared__ float smem[];
  float* band  = smem;                           // 16*4096 f32
  float* candv = band + 16 * 4096;               // 16*128 f32
  int*   candi = (int*)(candv + 16 * 128);       // 16*128 i32
  float* lmv   = (float*)(candi + 16 * 128);     // 16*16 f32
  int*   lmi   = (int*)(lmv + 256);              // 16*16 i32
  int*   winc  = (int*)(lmi + 256);              // 16 i32

  const int tid     = threadIdx.x;
  const int wave    = tid >> 5;
  const int lane    = tid & 31;
  const int rowbase = blockIdx.x * 16;

  // A operands (emb rows of this band) — identical for every column tile.
  const int am = lane & 15;          // A row within tile
  const int ak = (lane >> 4) * 2;    // K sub-lane offset
  v2f A0, A1;
  A0.x = emb[(rowbase + am) * DEMB + ak];
  A0.y = emb[(rowbase + am) * DEMB + ak + 1];
  A1.x = emb[(rowbase + am) * DEMB + 4 + ak];
  A1.y = emb[(rowbase + am) * DEMB + 4 + ak + 1];

  float sqr_[8];
  #pragma unroll
  for (int v = 0; v < 8; ++v)
    sqr_[v] = sq[rowbase + v + 8 * (lane >> 4)];

  for (int half = 0; half < 2; ++half) {
    // ---- build d2 band with WMMA ----
    for (int t = wave; t < 256; t += 8) {
      const int colbase = half * 4096 + t * 16;
      const int c = colbase + (lane & 15);       // B column = global col
      v2f B0, B1;
      B0.x = emb[c * DEMB + ak];
      B0.y = emb[c * DEMB + ak + 1];
      B1.x = emb[c * DEMB + 4 + ak];
      B1.y = emb[c * DEMB + 4 + ak + 1];

      v8f acc = {};
      acc = __builtin_amdgcn_wmma_f32_16x16x4_f32(false, A0, false, B0,
                                                  (short)0, acc, false, false);
      acc = __builtin_amdgcn_wmma_f32_16x16x4_f32(false, A1, false, B1,
                                                  (short)0, acc, false, false);
      const float sqc = sq[c];
      #pragma unroll
      for (int v = 0; v < 8; ++v) {
        const int m = v + 8 * (lane >> 4);
        float d2 = sqr_[v] + sqc - 2.f * acc[v];
        if (rowbase + m == c) d2 = INF_F;        // diagonal -> +inf
        band[m * 4096 + t * 16 + (lane & 15)] = d2;
      }
    }
    __syncthreads();

    // ---- top-64 within this half: lazy argmin ----
    const int row = tid >> 4;   // 0..15
    const int l16 = tid & 15;   // 16 threads per row
    float bv = INF_F; int bi = 0;
    for (int j = 0; j < 256; ++j) {
      const int c = l16 + (j << 4);
      const float v = band[row * 4096 + c];
      if (v < bv) { bv = v; bi = c; }
    }
    lmv[row * 16 + l16] = bv;
    lmi[row * 16 + l16] = bi;
    __syncthreads();

    for (int s = 0; s < 64; ++s) {
      if (l16 == 0) {
        float wv = INF_F; int wc = 0x7fffffff;
        for (int i = 0; i < 16; ++i) {
          const float v = lmv[row * 16 + i];
          const int   c = lmi[row * 16 + i];
          if (v < wv || (v == wv && c < wc)) { wv = v; wc = c; }
        }
        candv[row * 128 + half * 64 + s] = wv;
        candi[row * 128 + half * 64 + s] = half * 4096 + wc;
        winc[row] = wc;
      }
      __syncthreads();
      const int wc = winc[row];
      if (l16 == (wc & 15)) {
        band[row * 4096 + wc] = INF_F;
        float nv = INF_F; int ni = 0;
        for (int j = 0; j < 256; ++j) {
          const int c = l16 + (j << 4);
          const float v = band[row * 4096 + c];
          if (v < nv) { nv = v; ni = c; }
        }
        lmv[row * 16 + l16] = nv;
        lmi[row * 16 + l16] = ni;
      }
      __syncthreads();
    }
    __syncthreads();   // band reused next half
  }

  // ---- merge the two sorted 64-lists per row (sorted ascending, idx tiebreak)
  if (tid < 16) {
    const int row = tid;
    int ia = 0, ib = 0;
    for (int s = 0; s < 64; ++s) {
      const float va = candv[row * 128 + ia];
      const float vb = candv[row * 128 + 64 + ib];
      const int   ca = candi[row * 128 + ia];
      const int   cb = candi[row * 128 + 64 + ib];
      const bool takeA = (va < vb) || (va == vb && ca < cb);
      nbr[(long long)(rowbase + row) * KNBR + s] = takeA ? ca : cb;
      if (takeA) ++ia; else ++ib;
    }
  }
}

// ---------------------------------------------------------------------------
// Kernel C: edge features + MLP (28 -> 64 -> 1) via f32 WMMA + outputs.
// 8 waves x 16 edges per WG.
// ---------------------------------------------------------------------------
__global__ void edge_kernel(const float* __restrict__ x,
                            const int* __restrict__ pid,
                            const float* __restrict__ W1,
                            const float* __restrict__ b1,
                            const float* __restrict__ W2,
                            const float* __restrict__ b2,
                            const float* __restrict__ emb,
                            const int* __restrict__ nbr,
                            float* __restrict__ ef_out,
                            float* __restrict__ att_out,
                            float* __restrict__ dist_out,
                            int* __restrict__ y_out,
                            int* __restrict__ ei_out,
                            int* __restrict__ mask_out) {
  __shared__ float sW1[28 * 64];
  __shared__ float sb1[64];
  __shared__ float sW2[64];
  __shared__ float ef_s[8][16][28];
  __shared__ float h_s[8][16][64];
  __shared__ float dist_s[8][16];
  __shared__ float part_s[8][32];
  __shared__ int   src_s[8][16];
  __shared__ int   dst_s[8][16];

  const int tid  = threadIdx.x;
  const int wave = tid >> 5;
  const int lane = tid & 31;

  for (int i = tid; i < 28 * 64; i += 256) sW1[i] = W1[i];
  if (tid < 64) { sb1[tid] = b1[tid]; sW2[tid] = W2[tid]; }
  const float bb2 = b2[0];
  __syncthreads();

  const long long gbase = (long long)blockIdx.x * 128 + wave * 16;
  const int m   = lane & 15;
  const int hlf = lane >> 4;
  const long long g = gbase + m;
  const int dst = (int)(g >> 6);
  const int src = nbr[g];
  if (hlf == 0) { src_s[wave][m] = src; dst_s[wave][m] = dst; }

  // ef = [xs - xd, xs + xd]; lanes 0-15 do diff, 16-31 do sum
  for (int f = 0; f < NF; ++f) {
    const float xs = x[src * NF + f];
    const float xd = x[dst * NF + f];
    ef_s[wave][m][hlf * NF + f] = hlf ? (xs + xd) : (xs - xd);
  }
  if (hlf == 0) {
    float s2 = 0.f;
    #pragma unroll
    for (int d = 0; d < DEMB; ++d) {
      const float dv = emb[src * DEMB + d] - emb[dst * DEMB + d];
      s2 += dv * dv;
    }
    dist_s[wave][m] = __builtin_sqrtf(s2);
  }
  __syncthreads();

  // h(16x64) = relu(ef(16x28) @ W1(28x64) + b1), 4 N-tiles x 7 K-steps
  const int an  = lane & 15;
  const int ak2 = (lane >> 4) * 2;
  for (int nt = 0; nt < 4; ++nt) {
    const float bn = sb1[nt * 16 + an];
    v8f acc;
    #pragma unroll
    for (int v = 0; v < 8; ++v) acc[v] = bn;
    #pragma unroll
    for (int k0 = 0; k0 < 28; k0 += 4) {
      v2f A, B;
      A.x = ef_s[wave][an][k0 + ak2];
      A.y = ef_s[wave][an][k0 + ak2 + 1];
      B.x = sW1[(k0 + ak2)     * 64 + nt * 16 + an];
      B.y = sW1[(k0 + ak2 + 1) * 64 + nt * 16 + an];
      acc = __builtin_amdgcn_wmma_f32_16x16x4_f32(false, A, false, B,
                                                  (short)0, acc, false, false);
    }
    #pragma unroll
    for (int v = 0; v < 8; ++v) {
      const int mm = v + 8 * (lane >> 4);
      const float hv = acc[v];
      h_s[wave][mm][nt * 16 + an] = hv > 0.f ? hv : 0.f;
    }
  }
  __syncthreads();

  // att = h . W2 + b2 : two lanes per edge, 32 terms each
  float part = 0.f;
  for (int n = hlf * 32; n < hlf * 32 + 32; ++n)
    part += h_s[wave][m][n] * sW2[n];
  part_s[wave][lane] = part;
  __syncthreads();

  if (hlf == 0) {
    const float att  = part_s[wave][m] + part_s[wave][m + 16] + bb2;
    const float dist = dist_s[wave][m];
    const bool msk = (dist < 100.f) && (att > 0.f);   // MAX_RADIUS==EDGE_THRESHOLD
    const int s_ = src_s[wave][m];
    const int d_ = dst_s[wave][m];
    const int ps = pid[s_], pd = pid[d_];
    att_out[g]      = msk ? att : 0.f;
    dist_out[g]     = msk ? dist : 0.f;
    y_out[g]        = (msk && (ps == pd) && (ps > 0)) ? 1 : 0;
    ei_out[g]       = s_;        // edge_index row 0 = src
    ei_out[NK + g]  = d_;        // edge_index row 1 = dst
    mask_out[g]     = msk ? 1 : 0;
    dist_s[wave][m] = msk ? 1.f : 0.f;   // reuse as mask scale for ef writes
  }
  __syncthreads();

  const float mscale = dist_s[wave][m];
  for (int f = 0; f < NF; ++f)
    ef_out[g * 28 + hlf * NF + f] = ef_s[wave][m][hlf * NF + f] * mscale;
}

// ---------------------------------------------------------------------------
extern "C" void kernel_launch(void* const* d_in, const int* in_sizes, int n_in,
                              void* d_out, int out_size, void* d_ws, size_t ws_size,
                              hipStream_t stream) {
  const float* x   = (const float*)d_in[0];
  const int*   pid = (const int*)d_in[1];
  const float* W1  = (const float*)d_in[2];
  const float* b1  = (const float*)d_in[3];
  const float* W2  = (const float*)d_in[4];
  const float* b2  = (const float*)d_in[5];

  float* ws   = (float*)d_ws;
  float* sq   = ws;                       // NPTS
  float* embw = ws + NPTS;                // NPTS * DEMB
  int*   nbr  = (int*)(ws + NPTS + NPTS * DEMB);  // NK

  float* fo       = (float*)d_out;
  float* ef_out   = fo;                       // NK*28
  float* att_out  = fo + (size_t)NK * 28;     // NK
  float* dist_out = att_out + NK;             // NK
  int*   y_out    = (int*)(dist_out + NK);    // NK
  int*   ei_out   = y_out + NK;               // 2*NK
  int*   mask_out = ei_out + 2 * (size_t)NK;  // NK

  pack_kernel<<<(NPTS + 255) / 256, 256, 0, stream>>>(x, embw, sq);

  const int smem_bytes = (16 * 4096 + 16 * 128 * 2 + 256 * 2 + 16) * 4;
  (void)hipFuncSetAttribute((const void*)topk_kernel,
                            hipFuncAttributeMaxDynamicSharedMemorySize,
                            smem_bytes);
  topk_kernel<<<NPTS / 16, 256, smem_bytes, stream>>>(embw, sq, nbr);

  edge_kernel<<<NK / 128, 256, 0, stream>>>(x, pid, W1, b1, W2, b2, embw, nbr,
                                            ef_out, att_out, dist_out,
                                            y_out, ei_out, mask_out);
}